// HyperbolicDistanceAttention_83760452207347
// MI455X (gfx1250) — compile-verified
//
#include <hip/hip_runtime.h>
#include <cmath>

// Problem constants
#define B_   8
#define L_   1024
#define D_   768
#define H_   12
#define DH_  64
#define MAXN 0.99999f   // 1 - 1e-5
#define EPSF 1e-15f

typedef __attribute__((ext_vector_type(2))) float v2f;
typedef __attribute__((ext_vector_type(8))) float v8f;

// Probe for the gfx1250 async-to-LDS path (ASYNCcnt). Fallback keeps the
// synchronous staging.
#if defined(__has_builtin)
#if __has_builtin(__builtin_amdgcn_global_load_async_to_lds_b32) && \
    __has_builtin(__builtin_amdgcn_s_wait_asynccnt)
#define HAS_ASYNC_LDS 1
#endif
#endif

#ifdef HAS_ASYNC_LDS
// Builtin signature (from hipcc diagnostic): arg0 is address_space(1) int*
// (printed as "__device__ int *"), arg1 is the LDS address (address_space(3)).
// global generic pointer -> addrspace(1): identical 64-bit value
// LDS generic pointer    -> addrspace(3): low 32 bits are the LDS byte offset
#define TO_AS1(p) ((__attribute__((address_space(1))) int*)(unsigned long long)(uintptr_t)(p))
#define TO_AS3(p) ((__attribute__((address_space(3))) int*)(unsigned)(uintptr_t)(p))
#endif

__device__ __forceinline__ v8f zero8() {
  v8f z = {0.f,0.f,0.f,0.f,0.f,0.f,0.f,0.f};
  return z;
}

// full-precision fp32 WMMA: D(16x16) = A(16x4) * B(4x16) + C
__device__ __forceinline__ v8f wmma4(v2f a, v2f b, v8f c) {
  return __builtin_amdgcn_wmma_f32_16x16x4_f32(false, a, false, b, (short)0, c, false, false);
}

// artanh for x in [0, MAXN]
__device__ __forceinline__ float artanh_pos(float x) {
  return 0.5f * logf((1.f + x) / (1.f - x));
}

__device__ __forceinline__ float waveAllSum(float v) {
  #pragma unroll
  for (int off = 16; off > 0; off >>= 1) v += __shfl_xor(v, off, 32);
  return v;
}

// ---------------------------------------------------------------------------
// K0: row 2-norm (with _norm floor) of a (rows x 768) matrix. 384 thr = 12 waves.
// ---------------------------------------------------------------------------
__global__ __launch_bounds__(384) void krownorm(const float* __restrict__ x,
                                                float* __restrict__ xn) {
  int row = blockIdx.x;
  int wv = threadIdx.x >> 5, lane = threadIdx.x & 31;
  int e = wv * 64 + lane * 2;
  float a = x[(size_t)row * D_ + e];
  float b = x[(size_t)row * D_ + e + 1];
  float s = waveAllSum(a * a + b * b);
  __shared__ float hs[H_];
  if (lane == 0) hs[wv] = s;
  __syncthreads();
  if (threadIdx.x == 0) {
    float t = 0.f;
    for (int i = 0; i < H_; i++) t += hs[i];
    xn[row] = sqrtf(fmaxf(t, 1e-30f));
  }
}

// ---------------------------------------------------------------------------
// K1: C[8192x768] = A[8192x768] @ W^T, W is [768x768] row-major W[n][k].
// 256 thr = 8 waves; wave -> 16 rows x 64 cols (4 WMMA n-tiles).
// grid (8192/128, 768/64) = (64, 12)
// ---------------------------------------------------------------------------
__global__ __launch_bounds__(256) void kgemm_xwt(const float* __restrict__ A,
                                                 const float* __restrict__ W,
                                                 float* __restrict__ C) {
  int wv = threadIdx.x >> 5, lane = threadIdx.x & 31;
  int g = lane >> 4, li = lane & 15;
  int r0 = blockIdx.x * 128 + wv * 16;
  int c0 = blockIdx.y * 64;
  v8f acc[4];
  #pragma unroll
  for (int i = 0; i < 4; i++) acc[i] = zero8();
  const float* arow = A + (size_t)(r0 + li) * D_;
  for (int kk = 0; kk < D_; kk += 4) {
    v2f a; a.x = arow[kk + 2 * g]; a.y = arow[kk + 2 * g + 1];
    #pragma unroll
    for (int nt = 0; nt < 4; nt++) {
      int n = c0 + nt * 16 + li;
      v2f b; b.x = W[(size_t)n * D_ + kk + 2 * g]; b.y = W[(size_t)n * D_ + kk + 2 * g + 1];
      acc[nt] = wmma4(a, b, acc[nt]);
    }
  }
  #pragma unroll
  for (int nt = 0; nt < 4; nt++)
    #pragma unroll
    for (int i = 0; i < 8; i++)
      C[(size_t)(r0 + 8 * g + i) * D_ + c0 + nt * 16 + li] = acc[nt][i];
}

// ---------------------------------------------------------------------------
// K2: fused mobius_matvec scaling + project + beta_split (logmap0/expmap0).
// mx = x@W^T row; out in (B,H,L,DH) layout; aux = ||out_head||^2 = tanh(nh)^2.
// 384 thr = 12 waves = 12 heads, 2 elems/lane.
// ---------------------------------------------------------------------------
__global__ __launch_bounds__(384) void ksplit(const float* __restrict__ mx,
                                              const float* __restrict__ xn,
                                              float* __restrict__ outq,
                                              float* __restrict__ aux,
                                              float beta) {
  int row = blockIdx.x;             // b*L + l
  int b = row >> 10, l = row & (L_ - 1);
  int h = threadIdx.x >> 5, lane = threadIdx.x & 31;
  int e = h * 64 + lane * 2;
  float m0 = mx[(size_t)row * D_ + e];
  float m1 = mx[(size_t)row * D_ + e + 1];
  float hs = waveAllSum(m0 * m0 + m1 * m1);   // per-head sum of squares of mx
  __shared__ float hsum[H_];
  __shared__ float sS;
  if (lane == 0) hsum[h] = hs;
  __syncthreads();
  if (threadIdx.x == 0) {
    float tot = 0.f;
    for (int hh = 0; hh < H_; hh++) tot += hsum[hh];
    float mxn = sqrtf(fmaxf(tot, 1e-30f));
    float xv = xn[row];                               // >= 1e-15
    float t = tanhf(mxn / xv * artanh_pos(fminf(xv, MAXN)));
    float alpha = t / mxn;                            // mobius_matvec scale
    float nt = fabsf(t);                              // ||result||
    float pf = fminf(1.f, MAXN / fmaxf(nt, EPSF));    // project
    float n2 = nt * pf;
    // logmap0 scale * BETA_RATIO, all folded into one per-row scalar
    sS = artanh_pos(fminf(n2, MAXN)) / fmaxf(n2, EPSF) * pf * alpha * beta;
  }
  __syncthreads();
  float s = sS;
  float hn = sqrtf(fmaxf(s * s * hs, 1e-30f));        // per-head tangent norm
  float th = tanhf(hn);
  float f = th / hn * s;                              // expmap0 per head
  size_t ob = ((size_t)(b * H_ + h) * L_ + l) * DH_ + lane * 2;
  outq[ob]     = f * m0;
  outq[ob + 1] = f * m1;
  if (lane == 0) aux[(size_t)(b * H_ + h) * L_ + l] = th * th;  // ||head||^2
}

// exp(-scale * 2*artanh(t)) for one distance
__device__ __forceinline__ float attn_e(float qk, float q2v, float k2v, float scale) {
  float diff2 = fmaxf(q2v + k2v - 2.f * qk, 0.f);
  float den = fmaxf(1.f - 2.f * qk + q2v * k2v, EPSF);
  float t = sqrtf(fmaxf(diff2 / den, 0.f));
  t = fminf(t, MAXN);
  float dist = logf((1.f + t) / (1.f - t));   // 2 * artanh(t)
  return expf(-scale * dist);
}

// ---------------------------------------------------------------------------
// K3: fused hyperbolic-distance attention.
//   block = 16 q-rows x all 1024 k-cols for one (b,h).  256 thr = 8 waves.
//   Pass 1: WMMA qk -> e = exp(-scale*dist) -> row sums (no HBM round trip).
//   Pass 2: recompute qk, write attn with NT stores (write-only 402 MB stream,
//           keep L2 for q/k/v/nom), stage attn tile in LDS, WMMA partial of
//           nom = attn @ (lam*v) and den2 = attn @ (lam-1).
// grid (L/16, B*H) = (64, 96)
// ---------------------------------------------------------------------------
__global__ __launch_bounds__(256) void kattn(const float* __restrict__ q,
                                             const float* __restrict__ k,
                                             const float* __restrict__ v,
                                             const float* __restrict__ q2A,
                                             const float* __restrict__ k2A,
                                             const float* __restrict__ v2A,
                                             const float* __restrict__ scaleP,
                                             float* __restrict__ attnOut,
                                             float* __restrict__ nomA,
                                             float* __restrict__ den2A) {
  __shared__ float q2S[16], ssumS[16];
  __shared__ float k2S[L_], lamS[L_];
#ifdef HAS_ASYNC_LDS
  __shared__ float v2S[L_];
#endif
  __shared__ float redS[8][16];
  __shared__ float attnS[8][16 * 64];   // per-wave 16x64 attn tile (32 KB)
  __shared__ float nomS[16 * 64];

  int bh = blockIdx.y;
  int r0 = blockIdx.x * 16;
  int tid = threadIdx.x;
  int w = tid >> 5, lane = tid & 31, g = lane >> 4, li = lane & 15;
  const float scale = scaleP[0];
  const float* qrow  = q + ((size_t)bh * L_ + r0) * DH_;
  const float* kbase = k + (size_t)bh * L_ * DH_;
  const float* vbase = v + (size_t)bh * L_ * DH_;

  if (tid < 16) q2S[tid] = q2A[(size_t)bh * L_ + r0 + tid];

#ifdef HAS_ASYNC_LDS
  {
    const float* kg = k2A + (size_t)bh * L_;
    const float* vg = v2A + (size_t)bh * L_;
    for (int c = tid; c < L_; c += 256) {
      __builtin_amdgcn_global_load_async_to_lds_b32(TO_AS1(kg + c), TO_AS3(&k2S[c]), 0, 0);
      __builtin_amdgcn_global_load_async_to_lds_b32(TO_AS1(vg + c), TO_AS3(&v2S[c]), 0, 0);
    }
    __builtin_amdgcn_s_wait_asynccnt(0);
  }
  __syncthreads();   // all waves have waited on their own async loads
  for (int c = tid; c < L_; c += 256)
    lamS[c] = 2.f / fmaxf(1.f - v2S[c], EPSF);
#else
  for (int c = tid; c < L_; c += 256) {
    k2S[c] = k2A[(size_t)bh * L_ + c];
    float v2 = v2A[(size_t)bh * L_ + c];
    lamS[c] = 2.f / fmaxf(1.f - v2, EPSF);
  }
#endif
  for (int idx = tid; idx < 16 * 64; idx += 256) nomS[idx] = 0.f;
  __syncthreads();

  // ---- pass 1: row sums of e over this wave's 128 cols ----
  {
    int c0 = w * 128;
    float rs[8];
    #pragma unroll
    for (int i = 0; i < 8; i++) rs[i] = 0.f;
    v8f acc[8];
    #pragma unroll
    for (int i = 0; i < 8; i++) acc[i] = zero8();
    for (int kk = 0; kk < DH_; kk += 4) {
      v2f a; a.x = qrow[li * DH_ + kk + 2 * g]; a.y = qrow[li * DH_ + kk + 2 * g + 1];
      #pragma unroll
      for (int nt = 0; nt < 8; nt++) {
        int j = c0 + nt * 16 + li;
        v2f b; b.x = kbase[(size_t)j * DH_ + kk + 2 * g];
        b.y = kbase[(size_t)j * DH_ + kk + 2 * g + 1];
        acc[nt] = wmma4(a, b, acc[nt]);
      }
    }
    #pragma unroll
    for (int nt = 0; nt < 8; nt++)
      #pragma unroll
      for (int i = 0; i < 8; i++) {
        int rloc = 8 * g + i, j = c0 + nt * 16 + li;
        rs[i] += attn_e(acc[nt][i], q2S[rloc], k2S[j], scale);
      }
    #pragma unroll
    for (int i = 0; i < 8; i++) {
      #pragma unroll
      for (int off = 1; off < 16; off <<= 1) rs[i] += __shfl_xor(rs[i], off, 32);
    }
    if (li == 0) {
      #pragma unroll
      for (int i = 0; i < 8; i++) redS[w][8 * g + i] = rs[i];
    }
  }
  __syncthreads();
  if (tid < 16) {
    float s = 0.f;
    for (int ww = 0; ww < 8; ww++) s += redS[ww][tid];
    ssumS[tid] = s;
  }
  __syncthreads();

  // ---- pass 2: attn write + fused midpoint GEMM ----
  v8f acc2[4];
  #pragma unroll
  for (int i = 0; i < 4; i++) acc2[i] = zero8();
  float ds2[8];
  #pragma unroll
  for (int i = 0; i < 8; i++) ds2[i] = 0.f;

  for (int p = 0; p < 2; p++) {
    int c0 = w * 128 + p * 64;
    v8f acc[4];
    #pragma unroll
    for (int i = 0; i < 4; i++) acc[i] = zero8();
    for (int kk = 0; kk < DH_; kk += 4) {
      v2f a; a.x = qrow[li * DH_ + kk + 2 * g]; a.y = qrow[li * DH_ + kk + 2 * g + 1];
      #pragma unroll
      for (int nt = 0; nt < 4; nt++) {
        int j = c0 + nt * 16 + li;
        v2f b; b.x = kbase[(size_t)j * DH_ + kk + 2 * g];
        b.y = kbase[(size_t)j * DH_ + kk + 2 * g + 1];
        acc[nt] = wmma4(a, b, acc[nt]);
      }
    }
    #pragma unroll
    for (int nt = 0; nt < 4; nt++)
      #pragma unroll
      for (int i = 0; i < 8; i++) {
        int rloc = 8 * g + i, j = c0 + nt * 16 + li;
        float e = attn_e(acc[nt][i], q2S[rloc], k2S[j], scale);
        float at = e / ssumS[rloc];
        // non-temporal: 402 MB write-only stream, do not pollute L2
        __builtin_nontemporal_store(at, &attnOut[((size_t)bh * L_ + r0 + rloc) * L_ + j]);
        ds2[i] += at * (lamS[j] - 1.f);
        attnS[w][rloc * 64 + (nt * 16 + li)] = at;
      }
    // partial nom GEMM over this 64-col chunk: A = attn tile (LDS, per-wave,
    // DS ops are in-order within a wave so no barrier needed), B = lam_j * v_j
    for (int kk2 = 0; kk2 < 64; kk2 += 4) {
      int jl = kk2 + 2 * g;
      int jg = c0 + jl;
      v2f a2; a2.x = attnS[w][li * 64 + jl]; a2.y = attnS[w][li * 64 + jl + 1];
      float l0 = lamS[jg], l1 = lamS[jg + 1];
      const float* vp = vbase + (size_t)jg * DH_;
      #pragma unroll
      for (int nt2 = 0; nt2 < 4; nt2++) {
        int d = nt2 * 16 + li;
        v2f b2; b2.x = l0 * vp[d]; b2.y = l1 * vp[DH_ + d];
        acc2[nt2] = wmma4(a2, b2, acc2[nt2]);
      }
    }
  }

  // den2 cross-wave reduce
  #pragma unroll
  for (int i = 0; i < 8; i++) {
    #pragma unroll
    for (int off = 1; off < 16; off <<= 1) ds2[i] += __shfl_xor(ds2[i], off, 32);
  }
  if (li == 0) {
    #pragma unroll
    for (int i = 0; i < 8; i++) redS[w][8 * g + i] = ds2[i];
  }
  __syncthreads();
  if (tid < 16) {
    float s = 0.f;
    for (int ww = 0; ww < 8; ww++) s += redS[ww][tid];
    den2A[(size_t)bh * L_ + r0 + tid] = s;
  }
  __syncthreads();

  // nom cross-wave reduce (deterministic sequenced accumulation)
  for (int ww = 0; ww < 8; ww++) {
    if (w == ww) {
      #pragma unroll
      for (int nt2 = 0; nt2 < 4; nt2++)
        #pragma unroll
        for (int i = 0; i < 8; i++)
          nomS[(8 * g + i) * 64 + nt2 * 16 + li] += acc2[nt2][i];
    }
    __syncthreads();
  }
  for (int idx = tid; idx < 16 * 64; idx += 256) {
    int rloc = idx >> 6, d = idx & 63;
    nomA[((size_t)bh * L_ + r0 + rloc) * DH_ + d] = nomS[idx];
  }
}

// ---------------------------------------------------------------------------
// K4: weighted-midpoint finalize + beta_concat (logmap0 per head, concat,
// expmap0 full). Writes y2 (B,L,768) and y2n = ||y2 row||.
// 384 thr = 12 waves = heads.
// ---------------------------------------------------------------------------
__global__ __launch_bounds__(384) void kmid(const float* __restrict__ nomA,
                                            const float* __restrict__ den2A,
                                            float* __restrict__ y2,
                                            float* __restrict__ y2n,
                                            float invbeta) {
  int row = blockIdx.x;
  int b = row >> 10, l = row & (L_ - 1);
  int h = threadIdx.x >> 5, lane = threadIdx.x & 31;
  size_t bhq = (size_t)(b * H_ + h) * L_ + l;
  float n0 = nomA[bhq * DH_ + lane * 2];
  float n1 = nomA[bhq * DH_ + lane * 2 + 1];
  float dd = den2A[bhq];
  dd = (fabsf(dd) < 1e-10f) ? 1e-10f : dd;
  float t0 = n0 / dd, t1 = n1 / dd;                    // two_mean
  float s2 = waveAllSum(t0 * t0 + t1 * t1);
  float n = sqrtf(fmaxf(s2, 1e-30f));
  float tv = tanhf(0.5f * artanh_pos(fminf(n, MAXN))); // mobius_scalar_mul(0.5)
  float ny = fabsf(tv);
  float pf = fminf(1.f, MAXN / fmaxf(ny, EPSF));       // project
  float nmid = ny * pf;
  float art = artanh_pos(fminf(nmid, MAXN));
  float fu = art / fmaxf(nmid, EPSF) * pf * (tv / n) * invbeta; // applied to two_mean
  float a = art * invbeta;                             // head tangent norm
  __shared__ float hq[H_];
  __shared__ float feS;
  if (lane == 0) hq[h] = a * a;
  __syncthreads();
  if (threadIdx.x == 0) {
    float tot = 0.f;
    for (int hh = 0; hh < H_; hh++) tot += hq[hh];
    float un = sqrtf(fmaxf(tot, 1e-30f));
    float tu = tanhf(un);
    feS = tu / un;
    y2n[row] = fabsf(tu);
  }
  __syncthreads();
  float fe = feS;
  y2[(size_t)row * D_ + h * 64 + lane * 2]     = fe * fu * t0;
  y2[(size_t)row * D_ + h * 64 + lane * 2 + 1] = fe * fu * t1;
}

// ---------------------------------------------------------------------------
// K6: final mobius_matvec scale + project + mobius_add(bp) + project.
// All row stats reduce to scalars -> single pass. 384 thr.
// ---------------------------------------------------------------------------
__global__ __launch_bounds__(384) void kfinal(const float* __restrict__ mp,
                                              const float* __restrict__ y2n,
                                              const float* __restrict__ bp,
                                              float* __restrict__ yout) {
  int row = blockIdx.x;
  int wv = threadIdx.x >> 5, lane = threadIdx.x & 31;
  int e = wv * 64 + lane * 2;
  float m0 = mp[(size_t)row * D_ + e], m1 = mp[(size_t)row * D_ + e + 1];
  float b0 = bp[e], b1 = bp[e + 1];
  float smm = waveAllSum(m0 * m0 + m1 * m1);
  float sbb = waveAllSum(b0 * b0 + b1 * b1);
  float smb = waveAllSum(m0 * b0 + m1 * b1);
  __shared__ float r3[H_][3];
  __shared__ float cxS, cyS;
  if (lane == 0) { r3[wv][0] = smm; r3[wv][1] = sbb; r3[wv][2] = smb; }
  __syncthreads();
  if (threadIdx.x == 0) {
    float mm = 0.f, bb = 0.f, mb = 0.f;
    for (int hh = 0; hh < H_; hh++) { mm += r3[hh][0]; bb += r3[hh][1]; mb += r3[hh][2]; }
    float mpn = sqrtf(fmaxf(mm, 1e-30f));
    float xv = fmaxf(y2n[row], EPSF);
    float t = tanhf(mpn / xv * artanh_pos(fminf(xv, MAXN)));
    float bscale = t / mpn;
    float nm = fabsf(t);
    float pf = fminf(1.f, MAXN / fmaxf(nm, EPSF));
    bscale *= pf;                                      // m = bscale * mp
    float x2 = nm * pf; x2 *= x2;                      // ||m||^2
    float xy = bscale * mb;                            // <m, bp>
    float den = fmaxf(1.f + 2.f * xy + x2 * bb, EPSF);
    float cx = (1.f + 2.f * xy + bb) / den * bscale;   // coeff of mp
    float cy = (1.f - x2) / den;                       // coeff of bp
    // final project: ||cx*mp + cy*bp|| from scalars
    float on2 = cx * cx * mm + 2.f * cx * cy * mb + cy * cy * bb;
    float on = sqrtf(fmaxf(on2, 1e-30f));
    float pf2 = fminf(1.f, MAXN / on);
    cxS = cx * pf2; cyS = cy * pf2;
  }
  __syncthreads();
  yout[(size_t)row * D_ + e]     = cxS * m0 + cyS * b0;
  yout[(size_t)row * D_ + e + 1] = cxS * m1 + cyS * b1;
}

// ---------------------------------------------------------------------------
// Host launcher.  Workspace layout (floats), ~127.5 MB total:
//   mx(6291456) q(6291456) k(6291456) v(6291456) q2(98304) k2(98304)
//   v2(98304) nom(6291456) den2(98304) xn(8192) y2n(8192); y2 aliases q.
// ---------------------------------------------------------------------------
extern "C" void kernel_launch(void* const* d_in, const int* in_sizes, int n_in,
                              void* d_out, int out_size, void* d_ws, size_t ws_size,
                              hipStream_t stream) {
  const float* x      = (const float*)d_in[0];
  const float* Wq     = (const float*)d_in[1];
  const float* Wk     = (const float*)d_in[2];
  const float* Wv     = (const float*)d_in[3];
  const float* Wp     = (const float*)d_in[4];
  const float* bp     = (const float*)d_in[5];
  const float* scaleP = (const float*)d_in[6];

  float* yout    = (float*)d_out;
  float* attnOut = yout + (size_t)B_ * L_ * D_;

  const size_t NQ  = (size_t)B_ * H_ * L_ * DH_;  // 6291456 (== B*L*D)
  const size_t NBH = (size_t)B_ * H_ * L_;        // 98304
  const size_t NBL = (size_t)B_ * L_;             // 8192

  float* ws   = (float*)d_ws;
  float* mx   = ws;
  float* qb   = mx + NQ;
  float* kb   = qb + NQ;
  float* vb   = kb + NQ;
  float* q2   = vb + NQ;
  float* k2   = q2 + NBH;
  float* v2   = k2 + NBH;
  float* nom  = v2 + NBH;
  float* den2 = nom + NQ;
  float* xn   = den2 + NBH;
  float* y2n  = xn + NBL;
  float* y2   = qb;   // alias: q is dead after attention

  // BETA_RATIO = Beta(DH/2, 1/2) / Beta(D/2, 1/2)
  double lga = lgamma(32.0) + lgamma(0.5) - lgamma(32.5);
  double lgb = lgamma(384.0) + lgamma(0.5) - lgamma(384.5);
  float beta = (float)exp(lga - lgb);

  krownorm<<<dim3((unsigned)NBL), 384, 0, stream>>>(x, xn);

  kgemm_xwt<<<dim3(64, 12), 256, 0, stream>>>(x, Wq, mx);
  ksplit<<<dim3((unsigned)NBL), 384, 0, stream>>>(mx, xn, qb, q2, beta);
  kgemm_xwt<<<dim3(64, 12), 256, 0, stream>>>(x, Wk, mx);
  ksplit<<<dim3((unsigned)NBL), 384, 0, stream>>>(mx, xn, kb, k2, beta);
  kgemm_xwt<<<dim3(64, 12), 256, 0, stream>>>(x, Wv, mx);
  ksplit<<<dim3((unsigned)NBL), 384, 0, stream>>>(mx, xn, vb, v2, beta);

  kattn<<<dim3(L_ / 16, B_ * H_), 256, 0, stream>>>(qb, kb, vb, q2, k2, v2,
                                                    scaleP, attnOut, nom, den2);

  kmid<<<dim3((unsigned)NBL), 384, 0, stream>>>(nom, den2, y2, y2n, 1.0f / beta);

  kgemm_xwt<<<dim3(64, 12), 256, 0, stream>>>(y2, Wp, mx);

  kfinal<<<dim3((unsigned)NBL), 384, 0, stream>>>(mx, y2n, bp, yout);
}